// GeometryNoveltyAnalyzer_27333171871925
// MI455X (gfx1250) — compile-verified
//
#include <hip/hip_runtime.h>

// ---------------------------------------------------------------------------
// GeometryNoveltyAnalyzer for MI455X (gfx1250).
// Memory-bound problem: must stream depth (72.5MB) + conf (72.5MB) once.
// world_points is unused by the reference -> never read (saves 217MB).
// Pipeline:
//   1) pool28:     stream both maps, coalesced b128 loads, produce per-frame
//                  28x28 block sums (19x19-pixel base cells). 145MB traffic.
//   2) pool_grids: G_g = P_g * S28 * P_g^T via V_WMMA_F32_16X16X4_F32 chains
//                  (real matrix work on the CDNA5 matrix pipe).
//   3) pairwise:   64x8 pair novelty on pooled grids + max-reduce + isin mask.
// ---------------------------------------------------------------------------

typedef __attribute__((ext_vector_type(2))) float v2f;
typedef __attribute__((ext_vector_type(8))) float v8f;

#define F_FRAMES 64
#define H_DIM 532
#define W_DIM 532
#define S_SEL 8
#define CELL 19                         // base cell: 19x19 pixels
#define G28 28                          // base grid 28x28 (28*19 == 532)
#define NCELLS 261                      // 16 + 49 + 196
#define S28_STRIDE (G28 * G28)          // 784 floats per frame
#define S28_ARR (F_FRAMES * S28_STRIDE) // 50176 floats per array (d or c)
#define POOL_ARR (F_FRAMES * NCELLS)    // 16704 floats per array
#define CONF_THRESH 0.1f
#define EPS_NOV 1e-8f

// ---------------------------------------------------------------------------
// Kernel 1: per (frame, cell-row) block, sum 19 rows x 532 cols of depth and
// conf into 28 base-cell sums. All global loads are 16B (b128), fully
// coalesced; row base offsets are multiples of 532 (div by 4) -> 16B aligned.
// ---------------------------------------------------------------------------
__global__ __launch_bounds__(160)
void gna_pool28_kernel(const float* __restrict__ d, const float* __restrict__ c,
                       float* __restrict__ s28) {
    const int r = blockIdx.x;   // cell row 0..27
    const int f = blockIdx.y;   // frame 0..63
    const int t = threadIdx.x;  // 0..159 (133 active for load phase)

    __shared__ float4 cs4[133 * 2];     // column sums: [0..132]=d, [133..265]=c

    if (t < 133) {
        const size_t rowbase = ((size_t)f * H_DIM + (size_t)r * CELL) * W_DIM;
        const float4* d4 = (const float4*)(d + rowbase);
        const float4* c4 = (const float4*)(c + rowbase);
        float4 sd = make_float4(0.f, 0.f, 0.f, 0.f);
        float4 sc = make_float4(0.f, 0.f, 0.f, 0.f);
        for (int row = 0; row < CELL; ++row) {
            float4 vd = d4[(size_t)row * (W_DIM / 4) + t];
            float4 vc = c4[(size_t)row * (W_DIM / 4) + t];
            sd.x += vd.x; sd.y += vd.y; sd.z += vd.z; sd.w += vd.w;
            sc.x += vc.x; sc.y += vc.y; sc.z += vc.z; sc.w += vc.w;
        }
        cs4[t]       = sd;
        cs4[133 + t] = sc;
    }
    __syncthreads();

    if (t < G28) {
        const float* cd = (const float*)cs4;       // 532 d column sums
        const float* cc = cd + 133 * 4;            // 532 c column sums
        float sumd = 0.f, sumc = 0.f;
        #pragma unroll
        for (int k = 0; k < CELL; ++k) {
            sumd += cd[t * CELL + k];
            sumc += cc[t * CELL + k];
        }
        const size_t cellIdx = ((size_t)f * G28 + r) * G28 + t;
        s28[cellIdx]           = sumd;             // depth block sums
        s28[S28_ARR + cellIdx] = sumc;             // conf  block sums
    }
}

// ---------------------------------------------------------------------------
// Kernel 2: derive the 4x4 / 7x7 / 14x14 mean grids from the 28x28 sums via
// G = P * S * P^T with 0/1 pooling matrices, computed on the matrix pipe with
// V_WMMA_F32_16X16X4_F32 accumulation chains. One full wave (EXEC all 1s)
// per (frame, array). A-matrix 16x4 layout per ISA: lane M = l&15, holds
// K = {0,1} (lanes 0-15) / {2,3} (lanes 16-31); B uses the matching K pairs;
// C/D 16x16: lane N = l&15, M = vgpr + 8*(l>>4).
// ---------------------------------------------------------------------------
__global__ __launch_bounds__(32)
void gna_pool_grids_wmma_kernel(const float* __restrict__ s28,
                                float* __restrict__ pooled) {
    const int f   = blockIdx.x;   // frame
    const int arr = blockIdx.y;   // 0 = depth, 1 = conf
    const float* S = s28 + ((size_t)arr * F_FRAMES + f) * S28_STRIDE;
    float* out     = pooled + ((size_t)arr * F_FRAMES + f) * NCELLS;

    __shared__ float sS[S28_STRIDE];  // staged 28x28 sums
    __shared__ float sT[16 * G28];    // T = P*S, zero-padded to 16 rows

    const int lane  = threadIdx.x;    // 0..31, single wave
    const int n     = lane & 15;
    const int khalf = (lane >> 4) * 2;  // 0 or 2

    for (int idx = lane; idx < S28_STRIDE; idx += 32) sS[idx] = S[idx];
    __syncthreads();

    const int   gs[3]   = {4, 7, 14};
    const int   offs[3] = {0, 16, 65};
    const float inv[3]  = {1.f / (133.f * 133.f), 1.f / (76.f * 76.f),
                           1.f / (38.f * 38.f)};

    for (int gi = 0; gi < 3; ++gi) {
        const int g  = gs[gi];
        const int bs = G28 / g;        // 7, 4, 2 base cells per grid cell

        // ---- T = P * S : (g x 28), two 16-wide N tiles, K-loop of 7 ----
        for (int nt = 0; nt < 2; ++nt) {
            const int col = 16 * nt + n;
            v8f acc = {};
            for (int k4 = 0; k4 < 7; ++k4) {
                const int kc0 = 4 * k4 + khalf;
                v2f a, b;
                a.x = ((kc0 / bs) == n) ? 1.f : 0.f;         // P[n][kc0]
                a.y = (((kc0 + 1) / bs) == n) ? 1.f : 0.f;   // P[n][kc0+1]
                b.x = (col < G28) ? sS[kc0 * G28 + col] : 0.f;
                b.y = (col < G28) ? sS[(kc0 + 1) * G28 + col] : 0.f;
                acc = __builtin_amdgcn_wmma_f32_16x16x4_f32(
                          false, a, false, b, (short)0, acc, false, false);
            }
            #pragma unroll
            for (int v = 0; v < 8; ++v) {
                const int M = v + 8 * (lane >> 4);
                if (col < G28) sT[M * G28 + col] = acc[v];   // rows >= g are 0
            }
        }
        __syncthreads();

        // ---- G = T * P^T : (g x g), K-loop of 7 ----
        {
            v8f acc = {};
            for (int k4 = 0; k4 < 7; ++k4) {
                const int kc0 = 4 * k4 + khalf;
                v2f a, b;
                a.x = sT[n * G28 + kc0];                     // T[n][kc0]
                a.y = sT[n * G28 + kc0 + 1];
                b.x = ((kc0 / bs) == n) ? 1.f : 0.f;         // P^T[kc0][n]
                b.y = (((kc0 + 1) / bs) == n) ? 1.f : 0.f;
                acc = __builtin_amdgcn_wmma_f32_16x16x4_f32(
                          false, a, false, b, (short)0, acc, false, false);
            }
            #pragma unroll
            for (int v = 0; v < 8; ++v) {
                const int M = v + 8 * (lane >> 4);
                if (M < g && n < g)
                    out[offs[gi] + M * g + n] = acc[v] * inv[gi];
            }
        }
        __syncthreads();   // sT reused next scale
    }
}

// ---------------------------------------------------------------------------
// Kernel 3: pairwise novelty. One block, 512 threads = (frame i, selected j)
// pairs. Two passes per scale (cnt/dmax, then masked normalized mean), then
// max over j via LDS, isin() mask, write 64 outputs.
// ---------------------------------------------------------------------------
__global__ __launch_bounds__(512)
void gna_pairwise_kernel(const float* __restrict__ pooled,
                         const int* __restrict__ frame_idx,
                         const int* __restrict__ sel,
                         float* __restrict__ out) {
    const int t = threadIdx.x;      // 0..511
    const int i = t >> 3;           // frame 0..63
    const int j = t & 7;            // selected 0..7

    const int fi = frame_idx[i];
    const int fj = frame_idx[sel[j]];
    const float* di = pooled + (size_t)fi * NCELLS;
    const float* ci = pooled + (size_t)(F_FRAMES + fi) * NCELLS;
    const float* dj = pooled + (size_t)fj * NCELLS;
    const float* cj = pooled + (size_t)(F_FRAMES + fj) * NCELLS;

    const int gs[3]   = {4, 7, 14};
    const int offs[3] = {0, 16, 65};

    float novsum = 0.f;
    for (int gi = 0; gi < 3; ++gi) {
        const int cells = gs[gi] * gs[gi];
        const int o     = offs[gi];

        int   cnt  = 0;
        float dmax = -__builtin_inff();
        for (int k = 0; k < cells; ++k) {
            const float c1 = ci[o + k], c2 = cj[o + k];
            if (c1 > CONF_THRESH && c2 > CONF_THRESH) {
                ++cnt;
                dmax = fmaxf(dmax, fmaxf(di[o + k], dj[o + k]));
            }
        }
        const float dmax_safe = (cnt > 0) ? dmax : 1.0f;  // -inf <=> cnt==0
        const float rnorm = 1.0f / (dmax_safe + EPS_NOV);

        float s = 0.f;
        for (int k = 0; k < cells; ++k) {
            const float c1 = ci[o + k], c2 = cj[o + k];
            if (c1 > CONF_THRESH && c2 > CONF_THRESH) {
                const float diff = fabsf(di[o + k] - dj[o + k]) * (c1 + c2) * 0.5f;
                s += diff * rnorm;
            }
        }
        const float mean = s / (float)(cnt > 0 ? cnt : 1);
        const float nov  = fminf(fmaxf(mean, 0.f), 1.f);
        novsum += (cnt == 0) ? 1.f : nov;
    }
    const float frame_nov = novsum * (1.f / 3.f);

    __shared__ float red[512];
    red[t] = frame_nov;
    __syncthreads();

    if (j == 0) {
        float m = red[t];
        #pragma unroll
        for (int k = 1; k < S_SEL; ++k) m = fmaxf(m, red[t + k]);
        bool issel = false;
        #pragma unroll
        for (int k = 0; k < S_SEL; ++k) issel = issel || (fi == sel[k]);
        out[i] = issel ? 0.f : m;
    }
}

// ---------------------------------------------------------------------------
extern "C" void kernel_launch(void* const* d_in, const int* in_sizes, int n_in,
                              void* d_out, int out_size, void* d_ws, size_t ws_size,
                              hipStream_t stream) {
    (void)in_sizes; (void)n_in; (void)out_size; (void)ws_size;

    const float* depth = (const float*)d_in[0];  // (64,532,532,1) f32
    const float* conf  = (const float*)d_in[1];  // (64,532,532)   f32
    // d_in[2] = world_points: unused by the reference -> never read.
    const int* fidx = (const int*)d_in[3];       // (64) i32
    const int* sel  = (const int*)d_in[4];       // (8)  i32
    float* out = (float*)d_out;                  // (64) f32

    float* ws     = (float*)d_ws;
    float* s28    = ws;                    // 2 * 50176 floats (d,c 28x28 sums)
    float* pooled = ws + 2 * S28_ARR;      // 2 * 16704 floats (d,c 3-scale grids)
    // total workspace: 535,040 bytes

    dim3 g1(G28, F_FRAMES);                      // 28 cell-rows x 64 frames
    gna_pool28_kernel<<<g1, 160, 0, stream>>>(depth, conf, s28);

    dim3 g2(F_FRAMES, 2);                        // 64 frames x {depth, conf}
    gna_pool_grids_wmma_kernel<<<g2, 32, 0, stream>>>(s28, pooled);

    gna_pairwise_kernel<<<1, 512, 0, stream>>>(pooled, fidx, sel, out);
}